// EventEmulator_33440615367307
// MI455X (gfx1250) — compile-verified
//
#include <hip/hip_runtime.h>
#include <math.h>

// CDNA5 (gfx1250) wave32 WMMA types: 16x16x4 f32 -> A,B = 2 floats/lane, C/D = 8 floats/lane
typedef __attribute__((ext_vector_type(2))) float v2f;
typedef __attribute__((ext_vector_type(8))) float v8f;

#define NUM_BINS 5
#define NFRAMES  16
#define LINLOG_COEF 0.14978661367769956f  // log(20)/20

__device__ __forceinline__ float lin_log(float x) {
    // x <= 20 : linear ramp matching log at threshold; else natural log (f32, precise)
    return (x <= 20.0f) ? (x * LINLOG_COEF) : logf(fmaxf(x, 1e-9f));
}

__global__ void __launch_bounds__(256)
event_voxel_wmma(const float* __restrict__ frames,
                 const float* __restrict__ pos_thres,
                 const float* __restrict__ neg_thres,
                 float* __restrict__ out,
                 int HW, long nWaves)
{
    const int  lane = threadIdx.x & 31;
    const int  n    = lane & 15;   // N column (pixel-in-tile) and M row (bin) index
    const int  half = lane >> 4;   // selects K sub-rows per ISA A/B f32 layout
    const long wave = (long)blockIdx.x * (blockDim.x >> 5) + (threadIdx.x >> 5);
    if (wave >= nWaves) return;    // wave-uniform: EXEC stays all-1s for WMMA

    // Each wave owns 32 consecutive pixels; HW is a multiple of 32 so one batch b per wave.
    const long pixBase = wave * 32;
    const int  b   = (int)(pixBase / HW);
    const int  hw0 = (int)(pixBase - (long)b * HW);   // wave-uniform base
    const int  hw  = hw0 + lane;                      // this lane's own pixel

    const float pt = pos_thres[(long)b * HW + hw];
    const float nt = neg_thres[(long)b * HW + hw];
    const float* fptr = frames + ((long)b * NFRAMES) * HW + hw;

    // --- sequential event scan: one pixel per lane, registers only ---
    float cnt[NFRAMES];
    float base = lin_log(fptr[0]);
    #pragma unroll
    for (int f = 1; f < NFRAMES; ++f) {
        float lf   = lin_log(fptr[(long)f * HW]);
        float diff = lf - base;
        float pc   = floorf(fmaxf( diff, 0.0f) / pt);
        float nc   = floorf(fmaxf(-diff, 0.0f) / nt);
        base       = base + pc * pt - nc * nt;
        cnt[f - 1] = pc - nc;
    }
    cnt[NFRAMES - 1] = 0.0f;  // K=15 zero pad (also neutralizes its weight)

    // --- A = triangular bin weights (M=bin=n, K=frame): compile-time constants,
    //     half-dependence folded into single selects (no dynamic indexing) ---
    const float kf = (float)n;
    v2f aw[4];
    #pragma unroll
    for (int j = 0; j < 4; ++j) {
        // K rows held by this lane: c0 = 4j+2*half, c1 = c0+1 ; t(c) = (c+1)*4/15
        float t0 = (half ? (float)(4 * j + 3) : (float)(4 * j + 1)) * (4.0f / 15.0f);
        float t1 = (half ? (float)(4 * j + 4) : (float)(4 * j + 2)) * (4.0f / 15.0f);
        aw[j].x = fmaxf(0.0f, 1.0f - fabsf(t0 - kf));
        aw[j].y = fmaxf(0.0f, 1.0f - fabsf(t1 - kf));
    }

    // --- two D tiles: pixels [hw0..hw0+15] (owners = lanes 0-15) and
    //     [hw0+16..hw0+31] (owners = lanes 16-31); counts gathered via shuffles ---
    v8f acc0 = {0.f,0.f,0.f,0.f,0.f,0.f,0.f,0.f};
    v8f acc1 = {0.f,0.f,0.f,0.f,0.f,0.f,0.f,0.f};
    #pragma unroll
    for (int j = 0; j < 4; ++j) {
        // tile 0: pixel n owned by lane n
        float g0 = __shfl(cnt[4 * j + 0], n);
        float g1 = __shfl(cnt[4 * j + 1], n);
        float g2 = __shfl(cnt[4 * j + 2], n);
        float g3 = __shfl(cnt[4 * j + 3], n);
        v2f bm0; bm0.x = half ? g2 : g0; bm0.y = half ? g3 : g1;
        acc0 = __builtin_amdgcn_wmma_f32_16x16x4_f32(
                   false, aw[j], false, bm0, (short)0, acc0, false, false);

        // tile 1: pixel 16+n owned by lane 16+n
        float h0 = __shfl(cnt[4 * j + 0], 16 + n);
        float h1 = __shfl(cnt[4 * j + 1], 16 + n);
        float h2 = __shfl(cnt[4 * j + 2], 16 + n);
        float h3 = __shfl(cnt[4 * j + 3], 16 + n);
        v2f bm1; bm1.x = half ? h2 : h0; bm1.y = half ? h3 : h1;
        acc1 = __builtin_amdgcn_wmma_f32_16x16x4_f32(
                   false, aw[j], false, bm1, (short)0, acc1, false, false);
    }

    // D layout: lanes 0-15, VGPR r holds voxel[bin=r][pixel column n]; bins 5..15 zero.
    if (half == 0) {
        #pragma unroll
        for (int r = 0; r < NUM_BINS; ++r) {
            const long ob = ((long)b * NUM_BINS + r) * HW + hw0;
            out[ob + n]      = acc0[r];   // coalesced 64B row, tile 0
            out[ob + 16 + n] = acc1[r];   // coalesced 64B row, tile 1
        }
    }
}

extern "C" void kernel_launch(void* const* d_in, const int* in_sizes, int n_in,
                              void* d_out, int out_size, void* d_ws, size_t ws_size,
                              hipStream_t stream) {
    const float* frames = (const float*)d_in[0];
    // d_in[1] = t_frames: unused by the reference computation
    const float* pos    = (const float*)d_in[2];
    const float* neg    = (const float*)d_in[3];
    float*       out    = (float*)d_out;

    const int  B  = in_sizes[1] / 2;          // t_frames is (B, 2)
    const int  HW = in_sizes[2] / B;          // pos_thres is (B, 1, H, W)
    const long totalPixels = (long)B * HW;    // H*W = 512*512 -> divisible by 32
    const long nWaves = totalPixels / 32;     // 32 pixels per wave

    const int  threads = 256;                 // 8 waves per block (wave32)
    const int  wpb     = threads / 32;
    const long blocks  = (nWaves + wpb - 1) / wpb;

    event_voxel_wmma<<<(int)blocks, threads, 0, stream>>>(frames, pos, neg, out, HW, nWaves);
}